// FlowComposer_24696061952554
// MI455X (gfx1250) — compile-verified
//
#include <hip/hip_runtime.h>

// ---------------------------------------------------------------------------
// FlowComposer on MI455X (gfx1250, wave32).
//
// out[b,h,w,:] = project(Rᵀ_b · (depth[h,w]*pix_norm[h,w] - t)) flow, B=128.
// 478 MB f32 output -> ~20.5 us HBM-store floor @ 23.3 TB/s; inputs ~2 MB.
//
// Rotation offloaded to V_WMMA_F32_16X16X4_F32:
//   A(16x4)  = Rᵀ rows for 4 batches (rows 0-5 -> batches {0,1}; rows 8-13 ->
//              batches {2,3}; rows 6,7,14,15 zero) => D stays lane-local.
//   B(4x16)  = q-vectors for 16 pixels (lane n: qx,qy; lane n+16: qz,0).
// Each wave owns 32 pixels: B0 = even pixels (base+2n), B1 = odd pixels
// (base+2n+1), so lane n's two per-batch results are adjacent float2s and a
// single b128 store covers both. Per batch-group: 1 paired LDS A fetch
// (prefetched one group ahead), 2 back-to-back WMMAs, 2 b128 stores.
// Store addresses carried as a 32-bit byte offset off the uniform out base
// (total span 478 MB < 2^32) -> no 64-bit carry chains in the loop.
// Hot/tail split made via readfirstlane so it is a scalar branch.
// ---------------------------------------------------------------------------

#define HH 376
#define WW 1241
#define HWPIX (HH * WW)                 // 466616 (even)
#define FXc 718.856f
#define FYc 718.856f
#define CXc 607.1928f
#define CYc 185.2157f
#define WAVES_PER_BLOCK 8
#define PAIR_GROUPS ((HWPIX + 31) / 32) // 14582 (32-pixel wave groups)
#define FULL_PAIRS (HWPIX / 32)         // 14581 fully-populated groups
#define TAB_ENTRIES (33 * 32)           // 32 batch-groups + 1 zero pad group
#define BG_BYTE_STRIDE (4u * (unsigned)HWPIX * 8u)   // 14,931,712 B / group
#define BATCH_BYTE_STRIDE ((unsigned)HWPIX * 8u)     // 3,732,928 B (imm24 OK)

typedef __attribute__((ext_vector_type(2))) float v2f;
typedef __attribute__((ext_vector_type(8))) float v8f;

__device__ __forceinline__ void euler_to_R(float rx, float ry, float rz,
                                           float R[3][3]) {
  // R = Rz @ Ry @ Rx (matches reference)
  float cx = __cosf(rx), sx = __sinf(rx);
  float cy = __cosf(ry), sy = __sinf(ry);
  float cz = __cosf(rz), sz = __sinf(rz);
  R[0][0] = cz * cy;
  R[0][1] = cz * sy * sx - sz * cx;
  R[0][2] = sz * sx + cz * sy * cx;
  R[1][0] = sz * cy;
  R[1][1] = cz * cx + sz * sy * sx;
  R[1][2] = sz * sy * cx - cz * sx;
  R[2][0] = -sy;
  R[2][1] = cy * sx;
  R[2][2] = cy * cx;
}

__device__ __forceinline__ float2 projectf(float x, float y, float z,
                                           float ax, float ay) {
  const float FXW = FXc / (float)WW;
  const float FYH = FYc / (float)HH;
  float inv = __builtin_amdgcn_rcpf(z);
  inv = inv * (2.0f - z * inv);  // Newton refine
  return make_float2(fmaf(x * inv, FXW, ax), fmaf(y * inv, FYH, ay));
}

template <bool GUARD>
__device__ __forceinline__ void bg_loop(const float2 (&tabA)[TAB_ENTRIES],
                                        int lane, v2f Be, v2f Bo,
                                        float axe, float aye,
                                        float axo, float ayo,
                                        bool valid,
                                        char* __restrict__ outb,
                                        unsigned off) {
  float2 a = tabA[lane];  // prefetch bg=0
#pragma unroll 4
  for (int bg = 0; bg < 32; ++bg) {
    // Prefetch next A image (padded zero group makes bg+1 always legal)
    float2 an = tabA[(bg + 1) * 32 + lane];

    v2f A;
    A.x = a.x;
    A.y = a.y;
    v8f C = {0.f, 0.f, 0.f, 0.f, 0.f, 0.f, 0.f, 0.f};
    // Two independent WMMAs off one A fetch: 32 pixels x 4 batches
    v8f De = __builtin_amdgcn_wmma_f32_16x16x4_f32(
        false, A, false, Be, (short)0, C, false, false);  // even pixels
    v8f Do = __builtin_amdgcn_wmma_f32_16x16x4_f32(
        false, A, false, Bo, (short)0, C, false, false);  // odd pixels

    // lanes<16: batches bg*4+{0,1}; lanes>=16: bg*4+{2,3} (folded into off)
    float2 re0 = projectf(De[0], De[1], De[2], axe, aye);  // (bfirst,   even)
    float2 re1 = projectf(De[3], De[4], De[5], axe, aye);  // (bfirst+1, even)
    float2 ro0 = projectf(Do[0], Do[1], Do[2], axo, ayo);  // (bfirst,   odd)
    float2 ro1 = projectf(Do[3], Do[4], Do[5], axo, ayo);  // (bfirst+1, odd)

    if (!GUARD || valid) {
      // saddr(out) + zext(u32 voffset) global stores; 2nd folds to imm24
      *(float4*)(outb + off) =
          make_float4(re0.x, re0.y, ro0.x, ro0.y);
      *(float4*)(outb + off + BATCH_BYTE_STRIDE) =
          make_float4(re1.x, re1.y, ro1.x, ro1.y);
    }

    a = an;
    off += BG_BYTE_STRIDE;  // 32-bit add, no carry chain
  }
}

__global__ __launch_bounds__(256) void FlowComposer_24696061952554_kernel(
    const float* __restrict__ rot,    // (128,3)
    const float* __restrict__ depth,  // (H,W)
    const float* __restrict__ trans,  // (3,)
    float* __restrict__ out)          // (B,H,W,2)
{
  // ---- Stage 1: WMMA A-operand images for 32 batch-groups (+zero pad) ----
  // tabA[bg*32 + L]: L<16 -> (A[L][0], A[L][1]) ; L>=16 -> (A[L-16][2], 0)
  // A[m][k] = R_b[k][i],  m->(bb,i): m in [0,5]: bb=m/3, i=m%3 ;
  //                        m in [8,13]: bb=2+(m-8)/3, i=(m-8)%3 ; else 0.
  __shared__ float2 tabA[TAB_ENTRIES];  // 8.25 KB

  const int tid = threadIdx.x;
  for (int e = tid; e < TAB_ENTRIES; e += 256) {
    const int bg = e >> 5;
    const int L  = e & 31;
    const int m  = L & 15;
    int bb = 0, i = 0;
    bool valid_row = (bg < 32);
    if (m <= 5) {
      bb = m / 3;
      i  = m - 3 * bb;
    } else if (m >= 8 && m <= 13) {
      bb = 2 + (m - 8) / 3;
      i  = (m - 8) % 3;
    } else {
      valid_row = false;
    }
    float a0 = 0.0f, a1 = 0.0f;
    if (valid_row) {
      const int b = bg * 4 + bb;
      float R[3][3];
      euler_to_R(rot[3 * b + 0], rot[3 * b + 1], rot[3 * b + 2], R);
      if (L < 16) { a0 = R[0][i]; a1 = R[1][i]; }  // K=0, K=1
      else        { a0 = R[2][i]; a1 = 0.0f;    }  // K=2, K=3(=0)
    }
    tabA[e] = make_float2(a0, a1);
  }
  __syncthreads();

  // ---- Stage 2: per-wave 32-pixel group, loop over 32 batch-groups ----
  const int lane = tid & 31;
  const int wave = tid >> 5;
  const int pg   = blockIdx.x * WAVES_PER_BLOCK + wave;
  if (pg >= PAIR_GROUPS) return;  // whole-wave exit; EXEC full below

  const int psub = lane & 15;
  const int pe = pg * 32 + 2 * psub;  // even pixel (B0 column psub)
  // HWPIX even & pe even => pe valid <=> pe+1 valid
  const bool valid = (pe < HWPIX);
  const int pec = valid ? pe : (HWPIX - 2);  // clamped even index
  const int poc = pec + 1;

  const int he = pec / WW, we = pec - he * WW;
  const int ho = poc / WW, wo = poc - ho * WW;

  const float2 dpair = ((const float2*)depth)[pec >> 1];  // even+odd depths
  const float de = dpair.x, dod = dpair.y;
  const float tx = trans[0], ty = trans[1], tz = trans[2];

  const float pxe = ((float)we - CXc) * (1.0f / FXc);
  const float pye = ((float)he - CYc) * (1.0f / FYc);
  const float pxo = ((float)wo - CXc) * (1.0f / FXc);
  const float pyo = ((float)ho - CYc) * (1.0f / FYc);

  // B operands (q vectors), lane-local per assumed 4x16 layout
  v2f Be, Bo;
  if (lane < 16) {
    Be.x = de * pxe - tx;  Be.y = de * pye - ty;
    Bo.x = dod * pxo - tx; Bo.y = dod * pyo - ty;
  } else {
    Be.x = de - tz;  Be.y = 0.0f;
    Bo.x = dod - tz; Bo.y = 0.0f;
  }

  // Folded projection constants: flow_x = (x/z)*FX/W + (CX - w)/W
  const float axe = (CXc - (float)we) * (1.0f / (float)WW);
  const float aye = (CYc - (float)he) * (1.0f / (float)HH);
  const float axo = (CXc - (float)wo) * (1.0f / (float)WW);
  const float ayo = (CYc - (float)ho) * (1.0f / (float)HH);

  const int bsel = (lane < 16) ? 0 : 2;
  // 32-bit byte offset of (batch bsel, pixel pe): max < 478 MB < 2^32
  const unsigned off0 =
      (unsigned)bsel * BATCH_BYTE_STRIDE + (unsigned)pe * 8u;
  char* outb = (char*)out;

  // pg is wave-uniform: force a scalar compare + branch for hot/tail split
  if (__builtin_amdgcn_readfirstlane(pg) < FULL_PAIRS) {
    // Hot path: all 32 pixels valid, branch-free inner loop
    bg_loop<false>(tabA, lane, Be, Bo, axe, aye, axo, ayo, true, outb, off0);
  } else {
    // Single partial group: one per-lane flag guards both pixels
    bg_loop<true>(tabA, lane, Be, Bo, axe, aye, axo, ayo, valid, outb, off0);
  }
}

extern "C" void kernel_launch(void* const* d_in, const int* in_sizes, int n_in,
                              void* d_out, int out_size, void* d_ws, size_t ws_size,
                              hipStream_t stream) {
  const float* rot   = (const float*)d_in[0];  // (128,3) f32
  const float* depth = (const float*)d_in[1];  // (376,1241) f32
  const float* trans = (const float*)d_in[2];  // (3,) f32
  float* out = (float*)d_out;                  // (128,376,1241,2) f32

  const int blocks = (PAIR_GROUPS + WAVES_PER_BLOCK - 1) / WAVES_PER_BLOCK;  // 1823
  FlowComposer_24696061952554_kernel<<<blocks, 256, 0, stream>>>(rot, depth,
                                                                 trans, out);
}